// GATLayer_87840671138247
// MI455X (gfx1250) — compile-verified
//
#include <hip/hip_runtime.h>
#include <hip/hip_bf16.h>

// GAT layer for MI455X (gfx1250, wave32).
// Problem: B=2, T=12, N=2000, F=D=128, K=16.
//   hh = h @ W^T          (48000x128 @ 128x128, fp32)  -> V_WMMA_F32_16X16X4_F32
//   s_i/s_j = hh . a      fused into GEMM epilogue via shfl butterflies
//   softmax over K=16 neighbors, out = sum_k alpha_k * hh[adj[k], :]
// hh (24.6 MB) lives in workspace and stays L2-resident (192 MB L2), so the
// ~393 MB of logical gather traffic in kernel 2 is served from L2, not HBM.
// Roofline: ~75 MB of true HBM traffic -> ~3.2 us floor at 23.3 TB/s.

typedef __attribute__((ext_vector_type(2))) float v2f;
typedef __attribute__((ext_vector_type(8))) float v8f;
typedef int gv4i __attribute__((vector_size(16)));   // matches async-LDS builtin

#define BDIM   2
#define TDIM   12
#define NNODE  2000
#define FDIM   128           // = D
#define KNBR   16
#define MTOT   (BDIM * TDIM * NNODE)   // 48000 rows
#define HPAD   132           // padded LDS row stride for the h tile (floats)
#define WPST   288           // pair-row stride for W in LDS (== 32 mod 64 banks)

// ---------------------------------------------------------------------------
// Kernel 1: hh = h @ W^T via WMMA f32 16x16x4, fused s_i / s_j epilogue.
// Block = 256 threads = 8 waves; each wave -> 16 rows x 128 cols.
// Grid = MTOT/128 = 375 (exact; EXEC stays all-ones as WMMA requires).
// h tile (66 KB, padded) staged via GLOBAL_LOAD_ASYNC_TO_LDS_B128 (ASYNCcnt
// path, no VGPR round-trip); W staged pair-interleaved (72 KB) so every B
// fragment is a single conflict-free ds_load_b64 straight into an even-
// aligned VGPR pair (no repack movs). Steady-state loop: LDS + WMMA only.
// ---------------------------------------------------------------------------
__global__ void __launch_bounds__(256)
gat_gemm_scores(const float* __restrict__ h,   // [MTOT][128]
                const float* __restrict__ W,   // [128][128]
                const float* __restrict__ a,   // [256]
                float* __restrict__ hh,        // [MTOT][128]
                float* __restrict__ si,        // [MTOT]
                float* __restrict__ sj)        // [MTOT]
{
    // Pair-interleaved W: sWp[(f>>1)*WPST + 2*d + (f&1)] = W[d][f].
    // A b64 load at [(k>>1)*WPST + 2*d] yields (W[d][k], W[d][k+1]) adjacent.
    __shared__ float sWp[64 * WPST];      // 72 KB
    __shared__ float sh[128 * HPAD];      // 66 KB: h tile, padded stride

    const float* __restrict__ htile = h + (size_t)blockIdx.x * 128 * FDIM;

    // ---- stage h tile: 4096 float4 chunks, 16 per thread, all EXEC on ----
#if __has_builtin(__builtin_amdgcn_global_load_async_to_lds_b128)
    #pragma unroll
    for (int i = threadIdx.x; i < 4096; i += 256) {
        const int row = i >> 5;            // 32 float4 per 128-float row
        const int c4  = (i & 31) * 4;
        __builtin_amdgcn_global_load_async_to_lds_b128(
            (gv4i*)(htile + (size_t)row * FDIM + c4),
            (gv4i*)(sh + row * HPAD + c4),
            0, 0);
    }
#else
    for (int i = threadIdx.x; i < 4096; i += 256) {
        const int row = i >> 5;
        const int c4  = (i & 31) * 4;
        *reinterpret_cast<float4*>(sh + row * HPAD + c4) =
            *reinterpret_cast<const float4*>(htile + (size_t)row * FDIM + c4);
    }
#endif

    // ---- stage W pair-interleaved (overlaps the async h copy) ----
    for (int i = threadIdx.x * 4; i < FDIM * FDIM; i += 256 * 4) {
        const int d = i >> 7;              // row of W
        const int f = i & 127;             // col of W (multiple of 4)
        const float4 w4 = *reinterpret_cast<const float4*>(W + i);
        v2f p01; p01[0] = w4.x; p01[1] = w4.y;
        v2f p23; p23[0] = w4.z; p23[1] = w4.w;
        *reinterpret_cast<v2f*>(sWp + ((f >> 1) + 0) * WPST + 2 * d) = p01;
        *reinterpret_cast<v2f*>(sWp + ((f >> 1) + 1) * WPST + 2 * d) = p23;
    }

#if __has_builtin(__builtin_amdgcn_global_load_async_to_lds_b128)
  #if __has_builtin(__builtin_amdgcn_s_wait_asynccnt)
    __builtin_amdgcn_s_wait_asynccnt(0);
  #else
    asm volatile("s_wait_asynccnt 0" ::: "memory");
  #endif
#endif
    __syncthreads();

    const int lane = threadIdx.x & 31;
    const int wave = threadIdx.x >> 5;
    const int half = lane >> 4;      // 0: lanes 0-15, 1: lanes 16-31
    const int l    = lane & 15;
    const int m0   = blockIdx.x * 128 + wave * 16;

    // 8 accumulator tiles, each 16x16 f32 = 8 VGPRs (ISA C/D layout:
    // lane-half selects M in 0-7 vs 8-15, vector index r = M row, N = l).
    v8f acc[8] = {};

    // A fragment (16x4 f32, ISA layout): VGPR0 = K(2*half), VGPR1 = K(2*half+1),
    // row = m0 + l. LDS row stride 132 => banks (4l+c, 4l+c+1), halves offset
    // by 2 banks: 32 lanes x 2 banks = 64 distinct banks, conflict-free b64.
    const int arow = (wave * 16 + l) * HPAD + 2 * half;

    for (int kk = 0; kk < FDIM; kk += 4) {
        const v2f afrag = *reinterpret_cast<const v2f*>(sh + arow + kk);
        // B pair row for this half: k = kk + 2*half -> pair (kk>>1) + half.
        // Banks: half0 lanes -> {2l+c, 2l+c+1}; half1 shifted by WPST%64 = 32
        // banks: fully disjoint -> conflict-free b64 across the wave.
        const int brow = ((kk >> 1) + half) * WPST + 2 * l;
        #pragma unroll
        for (int t = 0; t < 8; ++t) {
            // bfrag = (W[t*16+l][kk+2h], W[t*16+l][kk+2h+1])
            const v2f bfrag =
                *reinterpret_cast<const v2f*>(sWp + brow + 32 * t);
            acc[t] = __builtin_amdgcn_wmma_f32_16x16x4_f32(
                false, afrag, false, bfrag, (short)0, acc[t], false, false);
        }
    }

    // Store hh: lane holds (M = m0 + half*8 + r, d = t*16 + l); 16 consecutive
    // lanes store 64 contiguous bytes per (t, r).
    #pragma unroll
    for (int t = 0; t < 8; ++t) {
        #pragma unroll
        for (int r = 0; r < 8; ++r) {
            hh[(size_t)(m0 + half * 8 + r) * FDIM + t * 16 + l] = acc[t][r];
        }
    }

    // Fused scores: s_i[m] = sum_d hh[m,d]*a[d], s_j[m] = sum_d hh[m,d]*a[128+d].
    float aL[8], aH[8];
    #pragma unroll
    for (int t = 0; t < 8; ++t) {
        aL[t] = a[t * 16 + l];
        aH[t] = a[FDIM + t * 16 + l];
    }
    #pragma unroll
    for (int r = 0; r < 8; ++r) {
        float pi = 0.f, pj = 0.f;
        #pragma unroll
        for (int t = 0; t < 8; ++t) {
            pi += acc[t][r] * aL[t];
            pj += acc[t][r] * aH[t];
        }
        // Butterfly over the 16-lane half (masks 1,2,4,8 stay inside a half).
        #pragma unroll
        for (int mask = 1; mask <= 8; mask <<= 1) {
            pi += __shfl_xor(pi, mask, 32);
            pj += __shfl_xor(pj, mask, 32);
        }
        if (l == 0) {
            const int m = m0 + half * 8 + r;
            si[m] = pi;
            sj[m] = pj;
        }
    }
}

// ---------------------------------------------------------------------------
// Kernel 2: per-(b,t,n) softmax over K=16 neighbors + weighted aggregation.
// One wave32 per node instance; 6000 blocks x 8 waves = 48000 exactly.
// Neighbor rows of hh are gathered from L2 (hh fits easily in 192 MB).
// ---------------------------------------------------------------------------
__global__ void __launch_bounds__(256)
gat_aggregate(const float* __restrict__ hh,      // [B*T][N][128]
              const float* __restrict__ si,      // [B*T][N]
              const float* __restrict__ sj,      // [B*T][N]
              const long long* __restrict__ adj, // [N][16] int64
              float* __restrict__ out)           // [B*T][N][128]
{
    const int lane = threadIdx.x & 31;
    const int wave = threadIdx.x >> 5;
    const int gw   = blockIdx.x * 8 + wave;   // 0 .. 47999
    const int n    = gw % NNODE;
    const int bt   = gw / NNODE;

    const float* __restrict__ hh_bt = hh + (size_t)bt * NNODE * FDIM;
    const float  s_self = si[(size_t)bt * NNODE + n];

    // Both 16-lane halves compute identical k = lane&15 work so the
    // xor-butterflies below reduce over the full neighbor set in each half.
    const int k   = lane & 15;
    const int nbr = (int)adj[(size_t)n * KNBR + k];
    float e = s_self + sj[(size_t)bt * NNODE + nbr];

    float mx = e;
    #pragma unroll
    for (int mask = 1; mask <= 8; mask <<= 1)
        mx = fmaxf(mx, __shfl_xor(mx, mask, 32));
    float ex = __expf(e - mx);
    float sum = ex;
    #pragma unroll
    for (int mask = 1; mask <= 8; mask <<= 1)
        sum += __shfl_xor(sum, mask, 32);
    const float alpha = ex / sum;

    // Each lane owns 4 consecutive d's (32 lanes * float4 = 128 = D).
    float4 accv = make_float4(0.f, 0.f, 0.f, 0.f);
    #pragma unroll
    for (int kk = 0; kk < KNBR; ++kk) {
        const int   idxk = __shfl(nbr,   kk, 32);
        const float ak   = __shfl(alpha, kk, 32);
        const float4 v = reinterpret_cast<const float4*>(
                             hh_bt + (size_t)idxk * FDIM)[lane];
        accv.x += ak * v.x;
        accv.y += ak * v.y;
        accv.z += ak * v.z;
        accv.w += ak * v.w;
    }
    reinterpret_cast<float4*>(out + (size_t)gw * FDIM)[lane] = accv;
}

// ---------------------------------------------------------------------------
extern "C" void kernel_launch(void* const* d_in, const int* in_sizes, int n_in,
                              void* d_out, int out_size, void* d_ws, size_t ws_size,
                              hipStream_t stream)
{
    const float*     h   = (const float*)d_in[0];      // (B,T,N,F) f32
    const float*     W   = (const float*)d_in[1];      // (D,F)     f32
    const float*     a   = (const float*)d_in[2];      // (2D,)     f32
    const long long* adj = (const long long*)d_in[3];  // (N,K)     i64
    float*           out = (float*)d_out;              // (B,T,N,D) f32

    // Workspace layout: hh [MTOT*128] | si [MTOT] | sj [MTOT]  (~25 MB)
    float* hh = (float*)d_ws;
    float* si = hh + (size_t)MTOT * FDIM;
    float* sj = si + MTOT;

    gat_gemm_scores<<<MTOT / 128, 256, 0, stream>>>(h, W, a, hh, si, sj);
    gat_aggregate  <<<MTOT / 8,   256, 0, stream>>>(hh, si, sj, adj, out);
}